// JacobianReg_19318762898225
// MI455X (gfx1250) — compile-verified
//
#include <hip/hip_runtime.h>
#include <hip/hip_bf16.h>
#include <math.h>

// ---------------------------------------------------------------------------
// JacobianReg on MI455X (gfx1250, wave32).
//   jac[b] = G[b] * W^T  =>  ||jac||_F^2 = tr(G M G^T),  M = W^T W (10x10)
// Heavy kernel = data(4096x3072) @ W(3072x10) via V_WMMA_F32_16X16X4_F32,
// staged through LDS with GLOBAL_LOAD_ASYNC_TO_LDS_B128 double-buffering.
// ---------------------------------------------------------------------------

typedef __attribute__((ext_vector_type(2))) float v2f;
typedef __attribute__((ext_vector_type(8))) float v8f;
typedef __attribute__((ext_vector_type(4))) int   v4i;

#define Bn 4096
#define Dn 3072
#define Cn 10
#define Mm 9                     // m = C-1
#define EPS_REG 0.1f
#define NUM_STAB 1e-6f
#define ALPHA (1.0f - (float)Cn * NUM_STAB)

#define WAVES 4                  // waves per block = K-split factor
#define BK 64                    // K-slab per LDS stage
#define TSTRIDE 68               // 64 + 4 pad floats: stride%64==4 -> conflict-free A reads
#define KCHUNK (Dn / WAVES)      // 768 columns per wave
#define SLABS (KCHUNK / BK)      // 12 slabs per wave
#define NTILES (Bn / 16)         // 256 tiles of 16 samples
#define WP_FLOATS ((Dn / 4) * 64)  // 49152: fragment-ready padded W

// --------------------- CDNA5 async-to-LDS staging path ---------------------
#if defined(__has_builtin)
#if __has_builtin(__builtin_amdgcn_global_load_async_to_lds_b128) && \
    __has_builtin(__builtin_amdgcn_s_wait_asynccnt)
#define USE_ASYNC_LDS 1
#endif
#endif

#if defined(USE_ASYNC_LDS)
__device__ __forceinline__ void async_cp16(void* gsrc, void* ldst) {
    // Builtin wants int4 pointers: AS(1) ("__device__") source, AS(3) dest.
    // reinterpret_cast can't change AS; C-style casts can, so cast pointee
    // first in generic AS, then convert address space.
    __builtin_amdgcn_global_load_async_to_lds_b128(
        (__attribute__((address_space(1))) v4i*)(v4i*)gsrc,
        (__attribute__((address_space(3))) v4i*)(v4i*)ldst,
        /*offset=*/0, /*cpol=*/0);
}
#endif

// Stage one 16 x 64 f32 slab: 8 x B128 per wave, two 256B row runs per instr.
__device__ __forceinline__ void stage_slab(const float* __restrict__ data,
                                           float* buf, int row0, int kb, int lane) {
    #pragma unroll
    for (int it = 0; it < 8; ++it) {
        const int f  = it * 32 + lane;        // float4 index 0..255
        const int r  = f >> 4;                // row 0..15
        const int c4 = (f & 15) << 2;         // col 0..60
        const float* src = data + (size_t)(row0 + r) * Dn + kb + c4;
        float*       dst = buf + r * TSTRIDE + c4;
#if defined(USE_ASYNC_LDS)
        async_cp16((void*)src, (void*)dst);
#else
        *(float4*)dst = *(const float4*)src;
#endif
    }
}

// 16 x V_WMMA_F32_16X16X4_F32 over one slab.
//   A (16x4): lanes 0-15 rows, regs K+0/K+1; lanes 16-31 regs K+2/K+3.
//   B (4x16): from fragment-ready Wp -> single coalesced b64 per lane.
__device__ __forceinline__ v8f compute_slab(const float* buf,
                                            const float* __restrict__ Wp,
                                            int kb, int mrow, int khalf, int lane,
                                            v8f acc) {
    #pragma unroll
    for (int kk = 0; kk < BK; kk += 4) {
        v2f a = *(const v2f*)(buf + mrow * TSTRIDE + kk + khalf);
        const int kq = (kb + kk) >> 2;
        v2f b = *(const v2f*)(Wp + (size_t)kq * 64 + lane * 2);
        acc = __builtin_amdgcn_wmma_f32_16x16x4_f32(
            false, a, false, b, (short)0, acc, false, false);
    }
    return acc;
}

// ---------------------------------------------------------------------------
// Kernel 0: fragment-ready padded W:
//   Wp[kq*64 + lane*2 + r] = W[4*kq + (lane>>4)*2 + r, lane&15]  (0 if N>=10)
// ---------------------------------------------------------------------------
__global__ __launch_bounds__(256)
void jacreg_prepw(const float* __restrict__ W, float* __restrict__ Wp) {
    const int idx = blockIdx.x * 256 + threadIdx.x;   // 0..49151
    const int kq  = idx >> 6;
    const int rem = idx & 63;
    const int l   = rem >> 1;
    const int r   = rem & 1;
    const int n   = l & 15;
    const int krow = (kq << 2) + ((l >> 4) << 1) + r;
    Wp[idx] = (n < Cn) ? W[krow * Cn + n] : 0.f;
}

// ---------------------------------------------------------------------------
// Kernel 1: M = W^T W  (10x10 = 100 entries, one block per entry)
// ---------------------------------------------------------------------------
__global__ __launch_bounds__(128)
void jacreg_wtw(const float* __restrict__ W, float* __restrict__ Mout) {
    const int pair = blockIdx.x;             // 0..99
    const int k = pair / Cn, l = pair % Cn;
    float acc = 0.f;
    for (int d = threadIdx.x; d < Dn; d += 128)
        acc += W[d * Cn + k] * W[d * Cn + l];
    __shared__ float red[128];
    red[threadIdx.x] = acc;
    __syncthreads();
    for (int s = 64; s > 0; s >>= 1) {
        if (threadIdx.x < s) red[threadIdx.x] += red[threadIdx.x + s];
        __syncthreads();
    }
    if (threadIdx.x == 0) Mout[pair] = red[0];
}

// ---------------------------------------------------------------------------
// Kernel 2: per-tile logits via WMMA + analytic Jacobian epilogue
// grid = NTILES blocks, 128 threads (4 waves, K-split over D)
// ---------------------------------------------------------------------------
__global__ __launch_bounds__(128)
void jacreg_main(const float* __restrict__ data, const float* __restrict__ Wp,
                 const float* __restrict__ Mmat, float* __restrict__ partials)
{
    __shared__ float lds_tile[WAVES * 2 * 16 * TSTRIDE];  // double-buffered slabs
    __shared__ float lds_z[WAVES * 256];                  // 16x16 partial logits/wave

    const int tid    = threadIdx.x;
    const int wave   = tid >> 5;
    const int lane   = tid & 31;
    const int tileId = blockIdx.x;
    const int row0   = tileId * 16;

    float* buf0 = lds_tile + wave * (2 * 16 * TSTRIDE);
    float* buf1 = buf0 + 16 * TSTRIDE;

    const int mrow  = lane & 15;
    const int khalf = (lane >> 4) << 1;      // 0 or 2

    v8f acc = {0.f, 0.f, 0.f, 0.f, 0.f, 0.f, 0.f, 0.f};

    const int kbeg = wave * KCHUNK;

    // software pipeline: stage i+1 async while computing i
    stage_slab(data, buf0, row0, kbeg, lane);
    float* cur = buf0;
    float* nxt = buf1;
    for (int i = 0; i < SLABS; ++i) {
        const int kb = kbeg + i * BK;
        const bool has_next = (i + 1 < SLABS);
        if (has_next) stage_slab(data, nxt, row0, kb + BK, lane);
#if defined(USE_ASYNC_LDS)
        // async loads retire in order: <=8 outstanding => current slab landed
        if (has_next) __builtin_amdgcn_s_wait_asynccnt(8);
        else          __builtin_amdgcn_s_wait_asynccnt(0);
#endif
        acc = compute_slab(cur, Wp, kb, mrow, khalf, lane, acc);
        float* t = cur; cur = nxt; nxt = t;
    }

    // Spill 16x16 accumulator to LDS per ISA C/D layout:
    //   VGPR v: lanes 0-15 -> (M=v, N=lane); lanes 16-31 -> (M=v+8, N=lane-16)
    {
        const int nCol = lane & 15;
        const int mOff = (lane >> 4) << 3;
        float* zb = lds_z + wave * 256;
        #pragma unroll
        for (int v = 0; v < 8; ++v)
            zb[(v + mOff) * 16 + nCol] = acc[v];
    }
    __syncthreads();

    float contrib = 0.f;
    if (wave == 0 && lane < 16) {
        // Combine K-split partial logits in a FIXED order (determinism).
        float z[Cn];
        #pragma unroll
        for (int k = 0; k < Cn; ++k) {
            float zz = 0.f;
            #pragma unroll
            for (int w = 0; w < WAVES; ++w)
                zz += lds_z[w * 256 + lane * 16 + k];
            z[k] = zz;
        }

        // softmax + stabilization
        float mx = z[0];
        #pragma unroll
        for (int k = 1; k < Cn; ++k) mx = fmaxf(mx, z[k]);
        float e[Cn]; float S = 0.f;
        #pragma unroll
        for (int k = 0; k < Cn; ++k) { e[k] = expf(z[k] - mx); S += e[k]; }
        const float inv = 1.f / S;
        float pr[Cn], p[Cn], s[Cn];
        #pragma unroll
        for (int k = 0; k < Cn; ++k) {
            pr[k] = e[k] * inv;
            p[k]  = pr[k] * ALPHA + NUM_STAB;
            s[k]  = sqrtf(p[k]);
        }
        const float sm = s[Mm];
        const float q  = 1.f - sm;
        const float um = ALPHA * pr[Mm] / (2.f * sm);

        // v = M*pr, t = pr^T M pr   (M loads are lane-uniform -> scalar loads)
        float v[Cn]; float t = 0.f;
        #pragma unroll
        for (int k = 0; k < Cn; ++k) {
            float a = 0.f;
            #pragma unroll
            for (int l = 0; l < Cn; ++l) a += Mmat[k * Cn + l] * pr[l];
            v[k] = a;
            t += pr[k] * a;
        }
        const float M99 = Mmat[Mm * Cn + Mm];

        // ||G W^T||_F^2 = sum_i G_i M G_i^T,  G_i = a_i e_i + b_i e_9 + c_i pr
        float fro = 0.f;
        #pragma unroll
        for (int i = 0; i < Mm; ++i) {
            const float ui = ALPHA * pr[i] / (2.f * s[i]);
            const float ai = 2.f * ui / q;
            const float bi = 2.f * s[i] * um / (q * q);
            const float ci = -(ai + bi);
            fro += ai * (ai * Mmat[i * Cn + i]
                         + 2.f * (bi * Mmat[i * Cn + Mm] + ci * v[i]))
                 + bi * (bi * M99 + 2.f * ci * v[Mm])
                 + ci * ci * t;
        }
        const float jn = sqrtf(fro);

        float ssum = 0.f;
        #pragma unroll
        for (int k = 0; k < Cn; ++k) ssum += s[k];
        float arg = ssum * 0.31622776601683794f;        // 1/sqrt(10)
        arg = fminf(1.f, fmaxf(-1.f, arg));
        const float delta = 2.f * acosf(arg);

        float psum = 0.f;
        #pragma unroll
        for (int i = 0; i < Mm; ++i) psum += p[i];
        const float rho = (2.f * q - psum) / q;

        const float x = jn - delta / (rho * EPS_REG);
        contrib = (x > 0.f) ? x : expm1f(x);            // ELU
    }

    if (wave == 0) {
        #pragma unroll
        for (int off = 16; off > 0; off >>= 1)
            contrib += __shfl_xor(contrib, off, 32);
        if (lane == 0) partials[tileId] = contrib;
    }
}

// ---------------------------------------------------------------------------
// Kernel 3: deterministic tree-reduce of 256 tile partials -> mean
// ---------------------------------------------------------------------------
__global__ __launch_bounds__(256)
void jacreg_finalize(const float* __restrict__ partials, float* __restrict__ out) {
    __shared__ float red[NTILES];
    const int tid = threadIdx.x;
    red[tid] = partials[tid];
    __syncthreads();
    for (int s = NTILES / 2; s > 0; s >>= 1) {
        if (tid < s) red[tid] += red[tid + s];
        __syncthreads();
    }
    if (tid == 0) out[0] = red[0] * (1.0f / (float)Bn);
}

// ---------------------------------------------------------------------------
extern "C" void kernel_launch(void* const* d_in, const int* in_sizes, int n_in,
                              void* d_out, int out_size, void* d_ws, size_t ws_size,
                              hipStream_t stream) {
    const float* data = (const float*)d_in[0];   // [4096, 3072] f32
    const float* W    = (const float*)d_in[1];   // [3072, 10]   f32
    float* out        = (float*)d_out;           // scalar f32
    float* ws         = (float*)d_ws;

    float* Mws      = ws;          // 100 floats: W^T W
    float* partials = ws + 128;    // 256 floats: per-tile sums
    float* Wp       = ws + 512;    // 49152 floats: fragment-ready padded W

    jacreg_prepw<<<WP_FLOATS / 256, 256, 0, stream>>>(W, Wp);
    jacreg_wtw<<<Cn * Cn, 128, 0, stream>>>(W, Mws);
    jacreg_main<<<NTILES, 128, 0, stream>>>(data, Wp, Mws, partials);
    jacreg_finalize<<<1, 256, 0, stream>>>(partials, out);
}